// QueryAndGroup_1468878815324
// MI455X (gfx1250) — compile-verified
//
#include <hip/hip_runtime.h>

// ---------------------------------------------------------------------------
// QueryAndGroup (PointNet++ ball query + grouping) for MI455X / gfx1250.
//   inputs : query_xyz (B,Q,3) f32, support_xyz (B,N,3) f32, features (B,C,N) f32
//   outputs: grouped_xyz (B,3,Q,32) f32 ++ grouped_features (B,C,Q,32) f32
// Kernel 1: wave32-tiled ball query. Per iteration: 32 support points loaded
//           coalesced (one per lane, software-pipelined double buffer), two
//           V_WMMA_F32_16X16X4_F32 for the -2*q.s distance terms, 16 ballots
//           -> per-query 32-bit in-radius masks, ctz append (ascending order),
//           wave-uniform early exit once all 16 queries have 32 hits.
// Kernel 2: streaming feature gather (output-bandwidth bound, ~140 MB).
// ---------------------------------------------------------------------------

typedef __attribute__((ext_vector_type(2))) float v2f;
typedef __attribute__((ext_vector_type(8))) float v8f;

#define BQ   4
#define NQ   4096     // queries per batch
#define NSUP 16384    // support points per batch
#define CF   64       // feature channels
#define SAMP 32       // nsample
#define R2   0.04f    // radius^2
#define WPB  4        // waves per block in kernel 1

__global__ __launch_bounds__(32 * WPB)
void ball_query_xyz_kernel(const float* __restrict__ query,    // (B,NQ,3)
                           const float* __restrict__ support,  // (B,NSUP,3)
                           float* __restrict__ out_xyz,        // (B,3,NQ,SAMP)
                           int* __restrict__ idx_ws)           // (B,NQ,SAMP)
{
    __shared__ int lds_idx[WPB][16][SAMP];

    const int wave = threadIdx.x >> 5;
    const int lane = threadIdx.x & 31;
    const int tile = blockIdx.x * WPB + wave;      // 16-query tile id
    const int b    = tile / (NQ / 16);
    const int q0   = (tile % (NQ / 16)) * 16;
    const int mq   = lane & 15;
    const bool lo  = (lane < 16);

    // ---- queries: lane m and lane m+16 load query m (one-time cost)
    const float* qp = query + ((size_t)b * NQ + q0 + mq) * 3;
    const float qx = qp[0], qy = qp[1], qz = qp[2];
    const float qn = qx * qx + qy * qy + qz * qz;

    // A operand (16x4 f32, ISA layout: V0 = K0|K2 per lane half, V1 = K1|K3),
    // pre-scaled by -2 so WMMA directly yields -2*q.s
    v2f A;
    A.x = lo ? (-2.0f * qx) : (-2.0f * qz);
    A.y = lo ? (-2.0f * qy) : 0.0f;

    // |q|^2 for the 8 rows this lane's D-VGPRs hold (rows r / r+8)
    const int moff = lo ? 0 : 8;
    float qn_r[8];
#pragma unroll
    for (int r = 0; r < 8; ++r) qn_r[r] = __shfl(qn, r + moff, 32);

    int cnt = 0;  // per-query hit count (valid on lanes 0..15)

    // ---- software-pipelined scan: 32 support points / iteration, one per lane
    const float* sp0 = support + ((size_t)b * NSUP + lane) * 3;
    float sx = sp0[0], sy = sp0[1], sz = sp0[2];

    for (int n0 = 0; n0 < NSUP; n0 += 32) {
        // issue next chunk's load before touching this chunk (hides latency)
        const int np = (n0 + 32 < NSUP) ? (n0 + 32) : n0;
        const float* spn = support + ((size_t)b * NSUP + np + lane) * 3;
        const float nx = spn[0], ny = spn[1], nz = spn[2];

        const float sn = sx * sx + sy * sy + sz * sz;

        // redistribute coords for the two B operands (4x16 each) via permutes
        const float z_lo  = __shfl(sz, mq, 32);       // z of point n0+mq
        const float x_hi  = __shfl(sx, mq + 16, 32);  // x of point n0+16+mq
        const float y_hi  = __shfl(sy, mq + 16, 32);
        const float sn_lo = __shfl(sn, mq, 32);
        const float sn_hi = __shfl(sn, mq + 16, 32);

        v2f B0, B1;                       // V0 = K0|K2, V1 = K1|K3
        B0.x = lo ? sx : z_lo;   B0.y = lo ? sy : 0.0f;    // cols n0..n0+15
        B1.x = lo ? x_hi : sz;   B1.y = lo ? y_hi : 0.0f;  // cols n0+16..n0+31
        const float sn0 = lo ? sn : sn_lo;   // |s|^2 of this lane's column
        const float sn1 = lo ? sn_hi : sn;

        v8f acc0 = {}, acc1 = {};
        acc0 = __builtin_amdgcn_wmma_f32_16x16x4_f32(
            false, A, false, B0, (short)0, acc0, false, false);
        acc1 = __builtin_amdgcn_wmma_f32_16x16x4_f32(
            false, A, false, B1, (short)0, acc1, false, false);

        // d2 = -2*q.s + |q|^2 + |s|^2 ; one ballot per D-VGPR per chunk:
        // rows r via bits 0..15, rows r+8 via bits 16..31
        unsigned int rowm0[8], rowm1[8];
#pragma unroll
        for (int r = 0; r < 8; ++r)
            rowm0[r] = (unsigned int)__ballot(acc0[r] + qn_r[r] + sn0 < R2);
#pragma unroll
        for (int r = 0; r < 8; ++r)
            rowm1[r] = (unsigned int)__ballot(acc1[r] + qn_r[r] + sn1 < R2);

        if (lane < 16) {
            const int rsel = mq & 7;  // uniform-value select chains
            unsigned int rm0 = rowm0[0], rm1 = rowm1[0];
            rm0 = (rsel == 1) ? rowm0[1] : rm0;  rm1 = (rsel == 1) ? rowm1[1] : rm1;
            rm0 = (rsel == 2) ? rowm0[2] : rm0;  rm1 = (rsel == 2) ? rowm1[2] : rm1;
            rm0 = (rsel == 3) ? rowm0[3] : rm0;  rm1 = (rsel == 3) ? rowm1[3] : rm1;
            rm0 = (rsel == 4) ? rowm0[4] : rm0;  rm1 = (rsel == 4) ? rowm1[4] : rm1;
            rm0 = (rsel == 5) ? rowm0[5] : rm0;  rm1 = (rsel == 5) ? rowm1[5] : rm1;
            rm0 = (rsel == 6) ? rowm0[6] : rm0;  rm1 = (rsel == 6) ? rowm1[6] : rm1;
            rm0 = (rsel == 7) ? rowm0[7] : rm0;  rm1 = (rsel == 7) ? rowm1[7] : rm1;
            const unsigned int c0 = (mq < 8) ? (rm0 & 0xFFFFu) : (rm0 >> 16);
            const unsigned int c1 = (mq < 8) ? (rm1 & 0xFFFFu) : (rm1 >> 16);
            unsigned int bits = c0 | (c1 << 16);  // bit i <=> point n0+i
            while (bits && cnt < SAMP) {
                const int n = __builtin_ctz(bits);
                bits &= bits - 1;
                lds_idx[wave][mq][cnt++] = n0 + n;  // ascending index order
            }
        }
        // early exit: with r=0.2 in unit cube, ~550 expected hits per query,
        // so all 16 queries typically fill after ~1k of 16k points
        if (__ballot((lane < 16) && (cnt < SAMP)) == 0) break;

        sx = nx; sy = ny; sz = nz;  // rotate double buffer
    }

    // pad remaining slots with first found index (or 0 if none) — matches ref
    if (lane < 16) {
        const int i0 = (cnt > 0) ? lds_idx[wave][mq][0] : 0;
        for (int s = cnt; s < SAMP; ++s) lds_idx[wave][mq][s] = i0;
    }
    __syncthreads();

    // write idx scratch (B,NQ,SAMP): 512 ints per tile, coalesced
    const size_t idx_base = ((size_t)b * NQ + q0) * SAMP;
#pragma unroll
    for (int i = 0; i < 16; ++i) {
        const int flat = i * 32 + lane;
        idx_ws[idx_base + flat] = lds_idx[wave][flat >> 5][flat & 31];
    }

    // grouped_xyz (B,3,NQ,SAMP): relative coordinates, 1536 floats per tile
    for (int i = 0; i < 48; ++i) {
        const int flat = i * 32 + lane;
        const int c   = flat / 512;
        const int rem = flat - c * 512;
        const int m = rem >> 5, s = rem & 31;
        const int id = lds_idx[wave][m][s];
        const float sv = support[((size_t)b * NSUP + id) * 3 + c];
        const float qv = query[((size_t)b * NQ + q0 + m) * 3 + c];
        out_xyz[(((size_t)b * 3 + c) * NQ + (q0 + m)) * SAMP + s] = sv - qv;
    }
}

__global__ __launch_bounds__(256)
void group_features_kernel(const float* __restrict__ feat,   // (B,CF,NSUP)
                           const int* __restrict__ idx_ws,   // (B,NQ,SAMP)
                           float* __restrict__ out_feat)     // (B,CF,NQ,SAMP)
{
    const int bq   = blockIdx.x;          // b*NQ + q
    const int b    = bq / NQ;
    const int q    = bq - b * NQ;
    const int lane = threadIdx.x & 31;    // = sample s
    const int wave = threadIdx.x >> 5;    // channel stripe 0..7

    const int id = idx_ws[(size_t)bq * SAMP + lane];  // coalesced
    const float* fb = feat + (size_t)b * CF * NSUP;

#pragma unroll
    for (int i = 0; i < CF / 8; ++i) {
        const int c = wave + i * 8;
        // store: 32 consecutive floats per wave (128B, coalesced)
        out_feat[(((size_t)b * CF + c) * NQ + q) * SAMP + lane] =
            fb[(size_t)c * NSUP + id];  // gather, L2-resident
    }
}

extern "C" void kernel_launch(void* const* d_in, const int* in_sizes, int n_in,
                              void* d_out, int out_size, void* d_ws, size_t ws_size,
                              hipStream_t stream) {
    const float* query   = (const float*)d_in[0];  // (B,NQ,3)
    const float* support = (const float*)d_in[1];  // (B,NSUP,3)
    const float* feat    = (const float*)d_in[2];  // (B,CF,NSUP)

    float* out_xyz  = (float*)d_out;
    float* out_feat = out_xyz + (size_t)BQ * 3 * NQ * SAMP;
    int*   idx_ws   = (int*)d_ws;  // B*NQ*SAMP ints = 2 MB

    const int tiles = BQ * (NQ / 16);  // 1024 wave-tiles
    ball_query_xyz_kernel<<<tiles / WPB, 32 * WPB, 0, stream>>>(
        query, support, out_xyz, idx_ws);

    group_features_kernel<<<BQ * NQ, 256, 0, stream>>>(feat, idx_ws, out_feat);
}